// LGConv_21852793602104
// MI455X (gfx1250) — compile-verified
//
#include <hip/hip_runtime.h>
#include <stddef.h>
#include <stdint.h>

// LightGCN propagation, K=2 hops, D=64 f32 features.
// Device-built CSR (incoming edges) -> TDM gather-mode SpMM: source rows are
// DMA-gathered into LDS by the Tensor Data Mover (tensor_load_to_lds,
// D#.gather_mode=1, 32-bit row indices), double-buffered so the DMA for the
// next 8-edge chunk overlaps accumulation of the current one.
// feat (25.6MB) is L2-resident on MI455X (192MB L2): all gathers hit L2.

#define THREADS         256
#define FEAT_D          64
#define SCAN_CHUNK      4
#define SCAN_SPAN       (THREADS * SCAN_CHUNK)   // 1024 elements per scan block
#define WAVES_PER_BLOCK 8                        // wave32: 256 threads = 8 waves
#define TDM_ROWS        8                        // rows per gather descriptor (32b idx)

typedef unsigned int v4u __attribute__((ext_vector_type(4)));
typedef int          v8i __attribute__((ext_vector_type(8)));
typedef int          v4i __attribute__((ext_vector_type(4)));

// ---------- CSR build ----------

__global__ void count_kernel(const int* __restrict__ dst, int E, int* __restrict__ cnt) {
  int i = blockIdx.x * blockDim.x + threadIdx.x;
  if (i < E) atomicAdd(&cnt[dst[i]], 1);
}

__global__ void norm_kernel(const int* __restrict__ cnt, float* __restrict__ nrm, int N) {
  int i = blockIdx.x * blockDim.x + threadIdx.x;
  if (i < N) nrm[i] = 1.0f / sqrtf(fmaxf((float)cnt[i], 1.0f));
}

__global__ void scan_block_kernel(const int* __restrict__ cnt, int* __restrict__ rowStart,
                                  int* __restrict__ blockSums, int N) {
  __shared__ int lds[THREADS];
  const int t    = threadIdx.x;
  const int base = blockIdx.x * SCAN_SPAN + t * SCAN_CHUNK;

  int v[SCAN_CHUNK];
  int sum = 0;
#pragma unroll
  for (int i = 0; i < SCAN_CHUNK; ++i) {
    int idx = base + i;
    v[i] = (idx < N) ? cnt[idx] : 0;
    sum += v[i];
  }
  lds[t] = sum;
  __syncthreads();
  for (int offs = 1; offs < THREADS; offs <<= 1) {
    int x = (t >= offs) ? lds[t - offs] : 0;
    __syncthreads();
    lds[t] += x;
    __syncthreads();
  }
  int excl = lds[t] - sum;
  if (t == THREADS - 1) blockSums[blockIdx.x] = lds[t];

  int run = excl;
#pragma unroll
  for (int i = 0; i < SCAN_CHUNK; ++i) {
    int idx = base + i;
    if (idx < N) rowStart[idx] = run;
    run += v[i];
  }
}

__global__ void scan_sums_kernel(int* __restrict__ blockSums, int nb) {
  if (threadIdx.x == 0 && blockIdx.x == 0) {
    int run = 0;
    for (int i = 0; i < nb; ++i) { int v = blockSums[i]; blockSums[i] = run; run += v; }
  }
}

__global__ void scan_add_kernel(int* __restrict__ rowStart, const int* __restrict__ blockSums, int N) {
  int i = blockIdx.x * blockDim.x + threadIdx.x;
  if (i < N) rowStart[i] += blockSums[i / SCAN_SPAN];
}

__global__ void fill_kernel(const int* __restrict__ src, const int* __restrict__ dst, int E,
                            const int* __restrict__ rowStart, int* __restrict__ cursor,
                            int* __restrict__ csrSrc) {
  int i = blockIdx.x * blockDim.x + threadIdx.x;
  if (i < E) {
    int d    = dst[i];
    int slot = atomicAdd(&cursor[d], 1);
    csrSrc[rowStart[d] + slot] = src[i];
  }
}

// ---------- TDM gather-mode descriptor issue ----------
// Gathers m (<=8) rows of 64 f32 from `in` (row r = in + idx[r]*64) into LDS
// at lds_addr, one tensor_load_to_lds per call (32-bit index gather mode).
__device__ __forceinline__ void tdm_gather_rows(const float* in, unsigned int lds_addr,
                                                const int* idx, int m, int N) {
  const unsigned long long ga = (unsigned long long)(uintptr_t)in;  // 57-bit base

  v4u g0;
  g0[0] = 1u              // count = 1 valid descriptor
        | (1u << 30)      // gather_index_size = 32-bit
        | (1u << 31);     // gather_mode = 1
  g0[1] = lds_addr;                                        // lds_addr
  g0[2] = (unsigned int)ga;                                // global_addr[31:0]
  g0[3] = ((unsigned int)(ga >> 32) & 0x01FFFFFFu)         // global_addr[56:32]
        | (2u << 30);                                      // type = 2 ("image")

  v8i g1;
  g1[0] = (2 << 16);                           // data_size = 2 (4 bytes)
  g1[1] = (FEAT_D & 0xFFFF) << 16;             // tensor_dim0[15:0] = 64
  g1[2] = ((N & 0xFFFF) << 16)                 // tensor_dim1[15:0] = N (low)
        | ((FEAT_D >> 16) & 0xFFFF);           // tensor_dim0[31:16] = 0
  g1[3] = ((FEAT_D & 0xFFFF) << 16)            // tile_dim0 = 64
        | ((N >> 16) & 0xFFFF);                // tensor_dim1[31:16]
  g1[4] = (m & 0xFFFF);                        // tile_dim1 = #valid indices
  g1[5] = FEAT_D;                              // tensor_dim0_stride[31:0] = 64
  g1[6] = 0;                                   // stride hi / dim1_stride (ignored)
  g1[7] = 0;

  v4i g2; g2[0] = idx[0]; g2[1] = idx[1]; g2[2] = idx[2]; g2[3] = idx[3];
  v4i g3; g3[0] = idx[4]; g3[1] = idx[5]; g3[2] = idx[6]; g3[3] = idx[7];

#if __clang_major__ >= 23
  v8i gpad = (v8i)0;
  __builtin_amdgcn_tensor_load_to_lds(g0, g1, g2, g3, gpad, 0);
#else
  __builtin_amdgcn_tensor_load_to_lds(g0, g1, g2, g3, 0);
#endif
}

// ---------- propagation hop: double-buffered TDM gather SpMM ----------
// One wave per destination node; two 2KB LDS staging buffers per wave. The
// DMA for chunk c+1 is issued before chunk c is consumed; s_wait_tensorcnt(1)
// exploits in-order completion of a wave's TDM ops to overlap DMA with the
// accumulate. Lane owns 2 floats of the 64-float row (ds_load_b64,
// conflict-free across 64 banks). srcSquared fuses hop1 post-scale with hop2
// pre-scale (norm^2); applyDstNorm applies norm[dst] at the final store.
__global__ void __launch_bounds__(THREADS)
hop_tdm_kernel(const float* __restrict__ in, float* __restrict__ out,
               const int* __restrict__ rowStart, const int* __restrict__ cnt,
               const int* __restrict__ csrSrc, const float* __restrict__ nrm,
               int N, int srcSquared, int applyDstNorm) {
  __shared__ float stage[WAVES_PER_BLOCK][2][TDM_ROWS * FEAT_D];  // 8 x 2 x 2KB = 32KB

  const int lane = threadIdx.x & 31;
  const int wave = threadIdx.x >> 5;
  const int node = blockIdx.x * WAVES_PER_BLOCK + wave;
  if (node >= N) return;

  const int beg = __builtin_amdgcn_readfirstlane(rowStart[node]);
  const int num = __builtin_amdgcn_readfirstlane(cnt[node]);

  const unsigned int lds0 = (unsigned int)(uintptr_t)(&stage[wave][0][0]);
  const unsigned int lds1 = (unsigned int)(uintptr_t)(&stage[wave][1][0]);

  float acc0 = 0.0f, acc1 = 0.0f;   // this lane's 2 output floats

  if (num > 0) {
    // Prologue: issue DMA for chunk 0 into buffer 0.
    int mCur = (num < TDM_ROWS) ? num : TDM_ROWS;
    int idxCur[TDM_ROWS];
    {
      int myIdx = (lane < mCur) ? csrSrc[beg + lane] : 0;
#pragma unroll
      for (int j = 0; j < TDM_ROWS; ++j)
        idxCur[j] = __builtin_amdgcn_readlane(myIdx, j);
    }
    tdm_gather_rows(in, lds0, idxCur, mCur, N);

    int bufCur = 0;
    for (int c = 0; c < num; c += TDM_ROWS) {
      const int cNext = c + TDM_ROWS;
      int mNext = 0;
      int idxNext[TDM_ROWS];
#pragma unroll
      for (int j = 0; j < TDM_ROWS; ++j) idxNext[j] = 0;

      if (cNext < num) {
        // Warm the chunk after next while the DMAs run.
        if (cNext + TDM_ROWS < num)
          __builtin_prefetch(&csrSrc[beg + cNext + TDM_ROWS], 0, 3);

        mNext = (num - cNext < TDM_ROWS) ? (num - cNext) : TDM_ROWS;
        int myIdx = (lane < mNext) ? csrSrc[beg + cNext + lane] : 0;
#pragma unroll
        for (int j = 0; j < TDM_ROWS; ++j)
          idxNext[j] = __builtin_amdgcn_readlane(myIdx, j);

        // Issue next gather into the other buffer, then wait only for the
        // older of the two in-flight TDM ops (in-order completion).
        tdm_gather_rows(in, bufCur ? lds0 : lds1, idxNext, mNext, N);
        __builtin_amdgcn_s_wait_tensorcnt(1);
      } else {
        __builtin_amdgcn_s_wait_tensorcnt(0);
      }

      // Consume chunk c from buffer bufCur.
      const int mC = (num - c < TDM_ROWS) ? (num - c) : TDM_ROWS;
#pragma unroll
      for (int j = 0; j < TDM_ROWS; ++j) {
        if (j >= mC) break;
        float w = nrm[idxCur[j]];
        if (srcSquared) w *= w;
        const float2 v = *(const float2*)&stage[wave][bufCur][j * FEAT_D + lane * 2];
        acc0 += w * v.x;
        acc1 += w * v.y;
      }

#pragma unroll
      for (int j = 0; j < TDM_ROWS; ++j) idxCur[j] = idxNext[j];
      bufCur ^= 1;
    }
  }

  if (applyDstNorm) {
    const float wn = nrm[node];
    acc0 *= wn;
    acc1 *= wn;
  }
  float2 r; r.x = acc0; r.y = acc1;
  *(float2*)&out[(size_t)node * FEAT_D + lane * 2] = r;   // coalesced b64 store
}

// ---------- launch ----------

extern "C" void kernel_launch(void* const* d_in, const int* in_sizes, int n_in,
                              void* d_out, int out_size, void* d_ws, size_t ws_size,
                              hipStream_t stream) {
  const float* feat = (const float*)d_in[0];
  const int*   src  = (const int*)d_in[1];
  const int*   dst  = (const int*)d_in[2];
  (void)n_in; (void)out_size; (void)ws_size;

  const int N  = in_sizes[0] / FEAT_D;
  const int E  = in_sizes[1];
  const int nb = (N + SCAN_SPAN - 1) / SCAN_SPAN;

  // Workspace layout (256B-aligned slices).
  char*  ws  = (char*)d_ws;
  size_t off = 0;
  auto take = [&](size_t bytes) -> char* {
    char* p = ws + off;
    off = (off + bytes + 255) & ~(size_t)255;
    return p;
  };
  int*   cnt       = (int*)  take((size_t)N * 4);
  int*   cursor    = (int*)  take((size_t)N * 4);
  int*   rowStart  = (int*)  take((size_t)N * 4);
  int*   blockSums = (int*)  take((size_t)nb * 4);
  float* nrm       = (float*)take((size_t)N * 4);
  int*   csrSrc    = (int*)  take((size_t)E * 4);
  float* B         = (float*)take((size_t)N * FEAT_D * 4);  // hop-1 raw segment sum

  // Deterministic per-call init (ws is not re-poisoned between replays).
  hipMemsetAsync(cnt,    0, (size_t)N * 4, stream);
  hipMemsetAsync(cursor, 0, (size_t)N * 4, stream);

  const int eBlocks = (E + THREADS - 1) / THREADS;
  const int nBlocks = (N + THREADS - 1) / THREADS;

  count_kernel     <<<eBlocks, THREADS, 0, stream>>>(dst, E, cnt);
  norm_kernel      <<<nBlocks, THREADS, 0, stream>>>(cnt, nrm, N);
  scan_block_kernel<<<nb,      THREADS, 0, stream>>>(cnt, rowStart, blockSums, N);
  scan_sums_kernel <<<1,       32,      0, stream>>>(blockSums, nb);
  scan_add_kernel  <<<nBlocks, THREADS, 0, stream>>>(rowStart, blockSums, N);
  fill_kernel      <<<eBlocks, THREADS, 0, stream>>>(src, dst, E, rowStart, cursor, csrSrc);

  const int hopBlocks = (N + WAVES_PER_BLOCK - 1) / WAVES_PER_BLOCK;
  // Hop 1: B = segsum(feat[src] * norm[src])            (post-scale deferred)
  hop_tdm_kernel<<<hopBlocks, THREADS, 0, stream>>>(feat, B, rowStart, cnt, csrSrc, nrm, N,
                                                    /*srcSquared=*/0, /*applyDstNorm=*/0);
  // Hop 2: out = norm[dst] * segsum(B[src] * norm[src]^2)
  hop_tdm_kernel<<<hopBlocks, THREADS, 0, stream>>>(B, (float*)d_out, rowStart, cnt, csrSrc, nrm, N,
                                                    /*srcSquared=*/1, /*applyDstNorm=*/1);
}